// MetaBaseline_13761075216822
// MI455X (gfx1250) — compile-verified
//
#include <hip/hip_runtime.h>
#include <hip/hip_bf16.h>

// ---------------- episode constants ----------------
#define BB     2
#define WAY    5
#define SHOT   5
#define TOPK   5
#define QN     30
#define DIM    64
#define HW     441
#define HWP    448          // HW padded to multiple of 16 (zero-filled)
#define S25    (WAY*SHOT)   // 25 support images per episode
#define NBLK   (BB*QN*WAY)  // 300 (b,q,w) blocks

typedef __attribute__((ext_vector_type(2))) float v2f;
typedef __attribute__((ext_vector_type(8))) float v8f;

#define NEG_INF (-3.402823466e38f)

// Keep t[0] >= t[1] >= ... >= t[4]; insert v if it beats the current minimum.
__device__ __forceinline__ void top5_insert(float t[TOPK], float v) {
  if (v <= t[4]) return;
  if (v > t[2]) {
    if (v > t[1]) {
      if (v > t[0]) { t[4]=t[3]; t[3]=t[2]; t[2]=t[1]; t[1]=t[0]; t[0]=v; }
      else          { t[4]=t[3]; t[3]=t[2]; t[2]=t[1]; t[1]=v; }
    } else          { t[4]=t[3]; t[3]=t[2]; t[2]=v; }
  } else {
    if (v > t[3])   { t[4]=t[3]; t[3]=v; }
    else              t[4]=v;
  }
}

__device__ __forceinline__ float wave_sum(float x) {
  #pragma unroll
  for (int off = 16; off; off >>= 1) x += __shfl_xor(x, off, 32);
  return x;
}

// ---------------------------------------------------------------------------
// K1: per-row (length HW) L2 normalization -> padded HWP rows, plus mean pool.
// One wave per row, 8 rows per 256-thread block.
// ---------------------------------------------------------------------------
__global__ void k_normalize_rows(const float* __restrict__ in,
                                 float* __restrict__ out,
                                 float* __restrict__ pool, int nrows) {
  const int wave = threadIdx.x >> 5, lane = threadIdx.x & 31;
  const int row  = blockIdx.x * 8 + wave;
  if (row >= nrows) return;
  const float* r = in + (size_t)row * HW;
  float vals[14];
  float sum = 0.f, ss = 0.f;
  #pragma unroll
  for (int it = 0; it < 14; ++it) {
    int h = lane + it * 32;
    float v = (h < HW) ? r[h] : 0.f;
    vals[it] = v; sum += v; ss += v * v;
  }
  sum = wave_sum(sum);
  ss  = wave_sum(ss);
  const float rn = rsqrtf(ss);
  float* o = out + (size_t)row * HWP;
  #pragma unroll
  for (int it = 0; it < 14; ++it) {
    int h = lane + it * 32;
    if (h < HW) o[h] = vals[it] * rn;
  }
  if (lane < HWP - HW) o[HW + lane] = 0.f;     // zero pad: inert in dot products
  if (lane == 0) pool[row] = sum * (1.0f / HW);
}

// ---------------------------------------------------------------------------
// K2: class prototypes + cosine logits. One block per episode b.
// ---------------------------------------------------------------------------
__global__ void k_cosine(const float* __restrict__ qpool,
                         const float* __restrict__ spool,
                         float* __restrict__ logits) {
  __shared__ float qp[QN][DIM];
  __shared__ float pr[WAY][DIM];
  __shared__ float inv[QN + WAY];
  const int b = blockIdx.x, t = threadIdx.x;
  for (int i = t; i < QN * DIM; i += 256) qp[i >> 6][i & 63] = qpool[b * QN * DIM + i];
  for (int i = t; i < WAY * DIM; i += 256) {
    int w = i >> 6, d = i & 63;
    float sm = 0.f;
    #pragma unroll
    for (int sh = 0; sh < SHOT; ++sh) sm += spool[b * S25 * DIM + (w * SHOT + sh) * DIM + d];
    pr[w][d] = sm * (1.0f / SHOT);
  }
  __syncthreads();
  const int wave = t >> 5, lane = t & 31;
  for (int r = wave; r < QN + WAY; r += 8) {
    const float* src = (r < QN) ? &qp[r][0] : &pr[r - QN][0];
    float a = src[lane], c = src[lane + 32];
    float ss = wave_sum(a * a + c * c);
    if (lane == 0) inv[r] = rsqrtf(ss);
  }
  __syncthreads();
  for (int i = t; i < QN * WAY; i += 256) {
    int q = i / WAY, w = i % WAY;
    float dot = 0.f;
    #pragma unroll 8
    for (int d = 0; d < DIM; ++d) dot += qp[q][d] * pr[w][d];
    logits[b * QN * WAY + q * WAY + w] = dot * inv[q] * inv[QN + w];
  }
}

// ---------------------------------------------------------------------------
// K3: channel-level similarity. One block per (b,q,w).
//   C[d=64][n=320] = qn[q,d,:] . sn[w,shot,d',:]  (K = HWP = 448)
//   top-5 per d row over n, summed over d and k.
// ---------------------------------------------------------------------------
__global__ void k_gemm1_topk(const float* __restrict__ qn,
                             const float* __restrict__ sn,
                             float* __restrict__ sim_l) {
  __shared__ float Cst[16][320];
  __shared__ float cand[16][16 * TOPK];
  __shared__ float rowsum[16];
  __shared__ float accTot;
  const int blk = blockIdx.x;
  const int b = blk / (QN * WAY), rem = blk % (QN * WAY);
  const int q = rem / WAY, w = rem % WAY;
  const int t = threadIdx.x, wave = t >> 5, lane = t & 31;
  const int mlan = lane & 15, ksel2 = (lane >> 4) * 2;
  if (t == 0) accTot = 0.f;
  __syncthreads();

  const float* Aq = qn + (size_t)((b * QN + q) * DIM) * HWP;
  const float* Sb = sn + (size_t)(b * S25 * DIM) * HWP;

  for (int i = 0; i < 4; ++i) {                       // M strips of 16 d-rows
    const float* Arow = Aq + (size_t)(i * 16 + mlan) * HWP + ksel2;
    for (int j = wave; j < 20; j += 8) {              // N tiles (uniform per wave)
      const int n = j * 16 + mlan;
      const int shot = n >> 6, dp = n & 63;
      const float* Brow = Sb + (size_t)((w * SHOT + shot) * DIM + dp) * HWP + ksel2;
      v8f acc = {};
      for (int k0 = 0; k0 < HWP; k0 += 4) {
        v2f av = *(const v2f*)(Arow + k0);
        v2f bv = *(const v2f*)(Brow + k0);
        acc = __builtin_amdgcn_wmma_f32_16x16x4_f32(false, av, false, bv,
                                                    (short)0, acc, false, false);
      }
      const int rbase = (lane < 16) ? 0 : 8;          // C layout: VGPR r -> M=r / M=r+8
      const int col = j * 16 + mlan;
      #pragma unroll
      for (int r = 0; r < 8; ++r) Cst[rbase + r][col] = acc[r];
    }
    __syncthreads();
    // per-row top-5: 16 threads/row, 20 candidates each
    const int row = t >> 4, sub = t & 15;
    float tp[TOPK] = {NEG_INF, NEG_INF, NEG_INF, NEG_INF, NEG_INF};
    #pragma unroll 4
    for (int c = 0; c < 20; ++c) top5_insert(tp, Cst[row][sub * 20 + c]);
    #pragma unroll
    for (int k = 0; k < TOPK; ++k) cand[row][sub * TOPK + k] = tp[k];
    __syncthreads();
    if (t < 16) {
      float m5[TOPK] = {NEG_INF, NEG_INF, NEG_INF, NEG_INF, NEG_INF};
      for (int c = 0; c < 16 * TOPK; ++c) top5_insert(m5, cand[t][c]);
      rowsum[t] = m5[0] + m5[1] + m5[2] + m5[3] + m5[4];
    }
    __syncthreads();
    if (t == 0) {
      float s = 0.f;
      #pragma unroll
      for (int r = 0; r < 16; ++r) s += rowsum[r];
      accTot += s;
    }
    __syncthreads();
  }
  if (t == 0) sim_l[blk] = accTot;
}

// ---------------------------------------------------------------------------
// K4: pixel-level similarity. One block per (b,q,w).
//   C[h=441(->448)][n=5*441(->5*448)] with K = DIM = 64.
//   Processed as 28 M-strips x 5 shot-chunks of 16x448; per-thread running
//   top-5 across chunks (padded rows/cols excluded from selection).
//   A fragments (16 x v2f = 32 VGPRs) are loaded ONCE per strip and reused
//   across all 5 shots x 28 N-tiles: 2 loads per WMMA instead of 4.
// ---------------------------------------------------------------------------
__global__ void k_gemm2_topk(const float* __restrict__ qn,
                             const float* __restrict__ sn,
                             float* __restrict__ sim_p) {
  __shared__ float Cst[16][HWP];
  __shared__ float cand[16][16 * TOPK];
  __shared__ float rowsum[16];
  __shared__ float accTot;
  const int blk = blockIdx.x;
  const int b = blk / (QN * WAY), rem = blk % (QN * WAY);
  const int q = rem / WAY, w = rem % WAY;
  const int t = threadIdx.x, wave = t >> 5, lane = t & 31;
  const int mlan = lane & 15, ksel2 = (lane >> 4) * 2;
  if (t == 0) accTot = 0.f;
  __syncthreads();

  const float* Aq = qn + (size_t)((b * QN + q) * DIM) * HWP;
  const float* Sw = sn + (size_t)((b * S25 + w * SHOT) * DIM) * HWP;
  const int row = t >> 4, sub = t & 15;

  for (int i = 0; i < 28; ++i) {                      // M strips (h rows)
    const int m = i * 16 + mlan;
    // Hoist A fragments for this strip: reused by 5 shots x 28 N-tiles.
    v2f afr[DIM / 4];
    #pragma unroll
    for (int ks = 0; ks < DIM / 4; ++ks) {
      const int kk = ks * 4 + ksel2;
      afr[ks] = v2f{ Aq[(size_t)kk * HWP + m], Aq[(size_t)(kk + 1) * HWP + m] };
    }
    float tp[TOPK] = {NEG_INF, NEG_INF, NEG_INF, NEG_INF, NEG_INF};
    for (int shot = 0; shot < SHOT; ++shot) {
      __syncthreads();                                // Cst free of prior readers
      const float* Bs = Sw + (size_t)(shot * DIM) * HWP;
      for (int j = wave; j < 28; j += 8) {            // N tiles within chunk
        const int col = j * 16 + mlan;
        v8f acc = {};
        #pragma unroll
        for (int ks = 0; ks < DIM / 4; ++ks) {
          const int kk = ks * 4 + ksel2;
          v2f bv = v2f{ Bs[(size_t)kk * HWP + col], Bs[(size_t)(kk + 1) * HWP + col] };
          acc = __builtin_amdgcn_wmma_f32_16x16x4_f32(false, afr[ks], false, bv,
                                                      (short)0, acc, false, false);
        }
        const int rbase = (lane < 16) ? 0 : 8;
        #pragma unroll
        for (int r = 0; r < 8; ++r) Cst[rbase + r][col] = acc[r];
      }
      __syncthreads();
      // running top-5: 28 cols per thread; skip padded cols (>= HW)
      #pragma unroll 4
      for (int c = 0; c < 28; ++c) {
        const int col = sub * 28 + c;
        if (col < HW) top5_insert(tp, Cst[row][col]);
      }
    }
    #pragma unroll
    for (int k = 0; k < TOPK; ++k) cand[row][sub * TOPK + k] = tp[k];
    __syncthreads();
    if (t < 16) {
      const int rg = i * 16 + t;                      // global h row
      float m5[TOPK] = {NEG_INF, NEG_INF, NEG_INF, NEG_INF, NEG_INF};
      for (int c = 0; c < 16 * TOPK; ++c) top5_insert(m5, cand[t][c]);
      rowsum[t] = (rg < HW) ? (m5[0] + m5[1] + m5[2] + m5[3] + m5[4]) : 0.f;
    }
    __syncthreads();
    if (t == 0) {
      float s = 0.f;
      #pragma unroll
      for (int r = 0; r < 16; ++r) s += rowsum[r];
      accTot += s;
    }
    __syncthreads();
  }
  if (t == 0) sim_p[blk] = accTot;
}

// ---------------------------------------------------------------------------
// K5: BatchNorm1d (training stats over Q) + dilated conv1d (k=3, d=5).
// One block per episode; one wave per BN column.
// ---------------------------------------------------------------------------
__global__ void k_bn_conv(const float* __restrict__ logits,
                          const float* __restrict__ sim_l,
                          const float* __restrict__ sim_p,
                          const float* __restrict__ gamma,
                          const float* __restrict__ beta,
                          const float* __restrict__ cw,
                          float* __restrict__ out) {
  __shared__ float fbn[QN][3 * WAY];
  const int b = blockIdx.x, t = threadIdx.x;
  const int wave = t >> 5, lane = t & 31;
  const int c = wave;                                  // 0..14 (block = 480 threads)
  const float* src = (c < WAY) ? (logits + b * QN * WAY)
                   : (c < 2 * WAY) ? (sim_l + b * QN * WAY)
                                   : (sim_p + b * QN * WAY);
  const int cc = c % WAY;
  float v = (lane < QN) ? src[lane * WAY + cc] : 0.f;
  const float mean = wave_sum(v) * (1.0f / QN);
  const float d = (lane < QN) ? (v - mean) : 0.f;
  const float var = wave_sum(d * d) * (1.0f / QN);     // biased, as in torch BN
  const float nrm = rsqrtf(var + 1e-5f);
  if (lane < QN) fbn[lane][c] = d * nrm * gamma[c] + beta[c];
  __syncthreads();
  const float w0 = cw[0], w1 = cw[1], w2 = cw[2];
  for (int i = t; i < QN * WAY; i += 480) {
    const int qq = i / WAY, j = i % WAY;
    out[b * QN * WAY + qq * WAY + j] =
        w0 * fbn[qq][j] + w1 * fbn[qq][j + WAY] + w2 * fbn[qq][j + 2 * WAY];
  }
}

// ---------------------------------------------------------------------------
extern "C" void kernel_launch(void* const* d_in, const int* in_sizes, int n_in,
                              void* d_out, int out_size, void* d_ws, size_t ws_size,
                              hipStream_t stream) {
  (void)in_sizes; (void)n_in; (void)out_size; (void)ws_size;
  const float* q      = (const float*)d_in[0];   // [B*Q, D, HW]
  const float* s      = (const float*)d_in[1];   // [B*25, D, HW]
  const float* gamma  = (const float*)d_in[2];   // [15]
  const float* beta   = (const float*)d_in[3];   // [15]
  const float* convw  = (const float*)d_in[4];   // [3]
  float* out = (float*)d_out;                    // [B, Q, WAY]

  float* ws = (float*)d_ws;
  float* qn     = ws;                                // B*Q*D*HWP
  float* sn     = qn + (size_t)BB * QN * DIM * HWP;  // B*25*D*HWP
  float* qpool  = sn + (size_t)BB * S25 * DIM * HWP; // B*Q*D
  float* spool  = qpool + (size_t)BB * QN * DIM;     // B*25*D
  float* logits = spool + (size_t)BB * S25 * DIM;    // 300
  float* sim_l  = logits + NBLK;                     // 300
  float* sim_p  = sim_l + NBLK;                      // 300

  const int qrows = BB * QN * DIM;   // 3840
  const int srows = BB * S25 * DIM;  // 3200
  k_normalize_rows<<<qrows / 8, 256, 0, stream>>>(q, qn, qpool, qrows);
  k_normalize_rows<<<srows / 8, 256, 0, stream>>>(s, sn, spool, srows);
  k_cosine<<<BB, 256, 0, stream>>>(qpool, spool, logits);
  k_gemm1_topk<<<NBLK, 256, 0, stream>>>(qn, sn, sim_l);
  k_gemm2_topk<<<NBLK, 256, 0, stream>>>(qn, sn, sim_p);
  k_bn_conv<<<BB, 480, 0, stream>>>(logits, sim_l, sim_p, gamma, beta, convw, out);
}